// LinearSequenceModel_47863115547570
// MI455X (gfx1250) — compile-verified
//
#include <hip/hip_runtime.h>
#include <hip/hip_bf16.h>

// Problem sizes (fixed by the reference)
#define Bz 64
#define Sz 2048
#define Fz 256
#define Cz 32

typedef float v2f __attribute__((ext_vector_type(2)));
typedef float v8f __attribute__((ext_vector_type(8)));

// ---------------------------------------------------------------------------
// Kernel 1: emissions = X @ W^T + b  via V_WMMA_F32_16X16X4_F32 (full fp32).
// Memory-bound (134 MB of X @ 23.3 TB/s), so fp32 WMMA is free and exact.
// Each wave: one 16-row M strip, BOTH 16-wide N tiles (so X is read once;
// W is 32 KB and lives in cache).
//
// ISA VGPR layouts (CDNA5 ISA):
//   A 16x4 f32 : lane = (k>=2)*16 + m, vgpr = k&1  -> float2 per lane / step
//   B 4x16 f32 : lane = (k>=2)*16 + n, vgpr = k&1  -> float2 per lane / step
//   C/D 16x16  : vgpr r -> row r (lanes 0-15) / row r+8 (lanes 16-31), n=lane&15
// ---------------------------------------------------------------------------
__global__ __launch_bounds__(256) void emis_gemm_wmma(
    const float* __restrict__ X, const float* __restrict__ W,
    const float* __restrict__ bias, float* __restrict__ E) {
  const int lane = threadIdx.x & 31;
  const int wave = threadIdx.x >> 5;
  const long m0 = ((long)blockIdx.x * 8 + wave) * 16;  // 16-row tile base
  const int r  = lane & 15;   // m (for A) / n (for B)
  const int kp = lane >> 4;   // selects K pair {0,1} vs {2,3}

  const float* Arow = X + (m0 + r) * Fz + 2 * kp;        // A: row m0+r
  const float* B0   = W + (long)r * Fz + 2 * kp;         // B tile 0: c = r
  const float* B1   = W + (long)(r + 16) * Fz + 2 * kp;  // B tile 1: c = r+16

  v8f acc0 = {};
  v8f acc1 = {};
#pragma unroll 8
  for (int k = 0; k < Fz; k += 4) {
    v2f a  = *(const v2f*)(Arow + k);
    v2f b0 = *(const v2f*)(B0 + k);
    v2f b1 = *(const v2f*)(B1 + k);
    acc0 = __builtin_amdgcn_wmma_f32_16x16x4_f32(
        /*neg_a=*/false, a, /*neg_b=*/false, b0,
        /*c_mod=*/(short)0, acc0, /*reuse_a=*/false, /*reuse_b=*/false);
    acc1 = __builtin_amdgcn_wmma_f32_16x16x4_f32(
        false, a, false, b1, (short)0, acc1, false, false);
  }

  // Store D + bias. acc[rr] in this lane is element (m0 + rr + 8*(lane>>4), n).
  const int n  = lane & 15;
  const int mo = (lane >> 4) * 8;
  const float bb0 = bias[n];
  const float bb1 = bias[n + 16];
#pragma unroll
  for (int rr = 0; rr < 8; ++rr) {
    const long m = m0 + rr + mo;
    E[m * Cz + n]      = acc0[rr] + bb0;
    E[m * Cz + n + 16] = acc1[rr] + bb1;
  }
}

// ---------------------------------------------------------------------------
// Kernel 2: Viterbi forward scan + in-LDS backtrack.
// One wave (32 lanes) per batch element; lane j carries alpha_j.
// Transitions column T[:,j] is held in 32 registers per lane.
// alpha_i broadcast via v_readlane_b32 (SGPR source into v_add_f32).
// Backpointers (2047 x 32 bytes = 65.5 KB) live entirely in LDS
// (dynamic LDS — CDNA5 allows 320 KB per workgroup).
// ---------------------------------------------------------------------------
__global__ __launch_bounds__(32) void viterbi_scan(
    const float* __restrict__ E, const float* __restrict__ T,
    const float* __restrict__ startT, const float* __restrict__ endT,
    float* __restrict__ out) {
  extern __shared__ unsigned char smem[];
  unsigned char* bp = smem;                                  // (Sz-1)*Cz bytes
  float* fin = (float*)(smem + ((((Sz - 1) * Cz) + 127) & ~127));

  const int j = threadIdx.x;      // class index for this lane
  const int b = blockIdx.x;       // batch element for this workgroup

  // Column j of the transition matrix -> registers (fully unrolled).
  float Tc[Cz];
#pragma unroll
  for (int i = 0; i < Cz; ++i) Tc[i] = T[i * Cz + j];

  const float* Eb = E + ((long)b * Sz) * Cz + j;

  float alpha = startT[j] + Eb[0];
  float enext = Eb[Cz];  // software-prefetch e[t] one step ahead

  for (int t = 1; t < Sz; ++t) {
    const float e = enext;
    if (t + 1 < Sz) enext = Eb[(long)(t + 1) * Cz];

    // best_j = max_i (alpha_i + T[i][j]); arg = argmax.
    float best;
    int arg = 0;
    {
      float a0 = __int_as_float(
          __builtin_amdgcn_readlane(__float_as_int(alpha), 0));
      best = a0 + Tc[0];
    }
#pragma unroll
    for (int i = 1; i < Cz; ++i) {
      float ai = __int_as_float(
          __builtin_amdgcn_readlane(__float_as_int(alpha), i));
      float s = ai + Tc[i];
      if (s > best) { best = s; arg = i; }   // v_cmp + 2x v_cndmask
    }
    alpha = best + e;
    bp[(t - 1) * Cz + j] = (unsigned char)arg;  // ds_store_b8
  }

  // Termination + argmax over final scores.
  fin[j] = alpha + endT[j];
  __syncthreads();

  if (j == 0) {
    float bs = fin[0];
    int tag = 0;
#pragma unroll
    for (int i = 1; i < Cz; ++i) {
      if (fin[i] > bs) { bs = fin[i]; tag = i; }
    }
    out[b] = bs;  // best_score[b]

    // Backtrack through LDS backpointers; path stored as float.
    float* path = out + Bz + (long)b * Sz;
    path[Sz - 1] = (float)tag;
    for (int t = Sz - 2; t >= 0; --t) {
      tag = bp[t * Cz + tag];
      path[t] = (float)tag;
    }
  }
}

// ---------------------------------------------------------------------------
// Launch: inputs are (X, W, b, transitions, start_transitions, end_transitions)
// d_ws holds the 16.8 MB emissions buffer. d_out = best_score[64] ++ path[64*2048].
// ---------------------------------------------------------------------------
extern "C" void kernel_launch(void* const* d_in, const int* in_sizes, int n_in,
                              void* d_out, int out_size, void* d_ws,
                              size_t ws_size, hipStream_t stream) {
  (void)in_sizes; (void)n_in; (void)out_size; (void)ws_size;
  const float* X  = (const float*)d_in[0];
  const float* W  = (const float*)d_in[1];
  const float* bv = (const float*)d_in[2];
  const float* T  = (const float*)d_in[3];
  const float* sT = (const float*)d_in[4];
  const float* eT = (const float*)d_in[5];
  float* E   = (float*)d_ws;
  float* out = (float*)d_out;

  // 131072 rows / 16 per wave / 8 waves per block = 1024 blocks.
  emis_gemm_wmma<<<1024, 256, 0, stream>>>(X, W, bv, E);

  const size_t sh = ((((Sz - 1) * Cz) + 127) & ~127) + Cz * sizeof(float);
  viterbi_scan<<<Bz, 32, sh, stream>>>(E, T, sT, eT, out);
}